// SubmanifoldConvolution_10934986735759
// MI455X (gfx1250) — compile-verified
//
#include <hip/hip_runtime.h>
#include <hip/hip_bf16.h>

// Submanifold sparse conv: gather -> per-offset 16x16 GEMM (WMMA f32 16x16x4) -> scatter-add.
// features [N,16] f32, weight [27,16,16] f32, bias [16] f32, rules_in/out [27,250000] i32.
// Output: [N,16] f32 = bias + scatter-add of per-rule contributions.

typedef __attribute__((ext_vector_type(2))) float v2f;
typedef __attribute__((ext_vector_type(8))) float v8f;

#define K_OFF        27
#define R_RULES      250000
#define TILES_PER_K  (R_RULES / 16)      // 15625 (exact)
#define BLOCKS_PER_K 128
#define WAVES_PER_BLOCK 8
#define WAVES_PER_K  (BLOCKS_PER_K * WAVES_PER_BLOCK)

__global__ __launch_bounds__(256)
void subm_init_bias(float* __restrict__ out, const float* __restrict__ bias, int total) {
    int i = blockIdx.x * blockDim.x + threadIdx.x;
    if (i < total) out[i] = bias[i & 15];
}

__global__ __launch_bounds__(256)
void subm_conv_wmma(const float* __restrict__ F,      // [N,16]
                    const float* __restrict__ W,      // [27,16,16]
                    const int*   __restrict__ rin,    // [27,R]
                    const int*   __restrict__ rout,   // [27,R]
                    float*       __restrict__ out)    // [N,16]
{
    const int lane = threadIdx.x & 31;
    const int wv   = threadIdx.x >> 5;
    const int k    = blockIdx.x / BLOCKS_PER_K;
    const int lw   = (blockIdx.x % BLOCKS_PER_K) * WAVES_PER_BLOCK + wv;
    const int hi   = lane >> 4;        // 0: lanes 0-15, 1: lanes 16-31
    const int col  = lane & 15;

    // --- Load B = weight[k] (16x16) in the 16x16x4 B layout, once per wave. ---
    // Chunk c covers Cin rows 4c..4c+3. VGPR0: row 4c+2*hi, VGPR1: row 4c+2*hi+1, col = lane&15.
    const float* Wk = W + k * 256;
    const int rb = 2 * hi;
    v2f b0, b1, b2, b3;
    b0.x = Wk[( 0 + rb) * 16 + col];  b0.y = Wk[( 1 + rb) * 16 + col];
    b1.x = Wk[( 4 + rb) * 16 + col];  b1.y = Wk[( 5 + rb) * 16 + col];
    b2.x = Wk[( 8 + rb) * 16 + col];  b2.y = Wk[( 9 + rb) * 16 + col];
    b3.x = Wk[(12 + rb) * 16 + col];  b3.y = Wk[(13 + rb) * 16 + col];

    const int* rin_k  = rin  + k * R_RULES;
    const int* rout_k = rout + k * R_RULES;

    for (int t = lw; t < TILES_PER_K; t += WAVES_PER_K) {
        const int r0 = t * 16;

        // Gather A: 16 feature rows. Lane holds row M = lane&15; needs cols {4c+2hi, 4c+2hi+1}.
        const int idx = rin_k[r0 + col];
        const float* frow = F + (long long)idx * 16 + 2 * hi;
        v2f a0 = *(const v2f*)(frow +  0);
        v2f a1 = *(const v2f*)(frow +  4);
        v2f a2 = *(const v2f*)(frow +  8);
        v2f a3 = *(const v2f*)(frow + 12);

        // 16x16x16 fp32 GEMM = 4 chained V_WMMA_F32_16X16X4_F32
        v8f acc = {};
        acc = __builtin_amdgcn_wmma_f32_16x16x4_f32(false, a0, false, b0, (short)0, acc, false, false);
        acc = __builtin_amdgcn_wmma_f32_16x16x4_f32(false, a1, false, b1, (short)0, acc, false, false);
        acc = __builtin_amdgcn_wmma_f32_16x16x4_f32(false, a2, false, b2, (short)0, acc, false, false);
        acc = __builtin_amdgcn_wmma_f32_16x16x4_f32(false, a3, false, b3, (short)0, acc, false, false);

        // Scatter-add: D VGPR j holds row (j + 8*hi), col = lane&15.
        const int ro = rout_k[r0 + col];   // lane holds rules_out for row lane&15
        #pragma unroll
        for (int j = 0; j < 8; ++j) {
            int src = j + (hi << 3);                 // lanes 0-15 need rout[j]; 16-31 need rout[j+8]
            int row = __shfl(ro, src, 32);
            float v = acc[j];
            __hip_atomic_fetch_add(out + (long long)row * 16 + col, v,
                                   __ATOMIC_RELAXED, __HIP_MEMORY_SCOPE_AGENT);
        }
    }
}

extern "C" void kernel_launch(void* const* d_in, const int* in_sizes, int n_in,
                              void* d_out, int out_size, void* d_ws, size_t ws_size,
                              hipStream_t stream) {
    const float* features = (const float*)d_in[0];   // [N,16]
    const float* weight   = (const float*)d_in[1];   // [27,16,16]
    const float* bias     = (const float*)d_in[2];   // [16]
    const int*   rules_in = (const int*)d_in[3];     // [27,R]
    const int*   rules_out= (const int*)d_in[4];     // [27,R]
    float* out = (float*)d_out;

    const int total = out_size;                      // N*16
    subm_init_bias<<<(total + 255) / 256, 256, 0, stream>>>(out, bias, total);

    dim3 grid(K_OFF * BLOCKS_PER_K);
    subm_conv_wmma<<<grid, 256, 0, stream>>>(features, weight, rules_in, rules_out, out);
}